// InfoGCN_75299366633587
// MI455X (gfx1250) — compile-verified
//
#include <hip/hip_runtime.h>

typedef float v2f __attribute__((ext_vector_type(2)));
typedef float v8f __attribute__((ext_vector_type(8)));

#define TT 64     // T
#define CCH 64    // C
#define VV 25     // V
#define VP 32     // padded V
#define BBATCH 64 // B
#define NSTEP 4

__device__ __forceinline__ float lrelu(float x) { return x > 0.0f ? x : 0.01f * x; }

// out[c,v] = sum_u X[c,u] * A[v,u]   (X: 64xVP in LDS; A^T B-fragments cached in regs aF[7])
// Wave handles tile (mo: c-rows, no: v-cols). K = u padded to 28; padding zeros live in aF.
__device__ __forceinline__ void amm(const float* X, const v2f* aF, float* D,
                                    int mo, int no, int lane, int koff)
{
    v8f acc = {};
    const int Mrow = mo * 16 + (lane & 15);
    const int Ncol = no * 16 + (lane & 15);
#pragma unroll
    for (int k7 = 0; k7 < 7; ++k7) {
        const int kk = 4 * k7 + koff;
        v2f a; a.x = X[Mrow * VP + kk]; a.y = X[Mrow * VP + kk + 1];
        acc = __builtin_amdgcn_wmma_f32_16x16x4_f32(false, a, false, aF[k7], (short)0, acc,
                                                    false, false);
    }
    const int radd = (lane >> 4) * 8;
#pragma unroll
    for (int j = 0; j < 8; ++j)
        D[(mo * 16 + radd + j) * VP + Ncol] = acc[j];
}

// out[o,v] = sum_c W[o,c] * X[c,v] + bias[o]  (W prefetched as register fragments wf[16])
__device__ __forceinline__ void cmm(const v2f* wf, const float* X, float* D,
                                    const float* bias, bool relu,
                                    int mo, int no, int lane, int koff)
{
    v8f acc = {};
    const int Ncol = no * 16 + (lane & 15);
#pragma unroll
    for (int j = 0; j < 16; ++j) {
        const int kk = 4 * j + koff;
        v2f bb; bb.x = X[kk * VP + Ncol]; bb.y = X[(kk + 1) * VP + Ncol];
        acc = __builtin_amdgcn_wmma_f32_16x16x4_f32(false, wf[j], false, bb, (short)0, acc,
                                                    false, false);
    }
    const int radd = (lane >> 4) * 8;
#pragma unroll
    for (int j = 0; j < 8; ++j) {
        const int row = mo * 16 + radd + j;
        float v = acc[j] + bias[row];
        if (relu) v = lrelu(v);
        D[row * VP + Ncol] = v;
    }
}

__global__ __launch_bounds__(256, 1)
void infogcn_rk4_kernel(const float* __restrict__ z0, const float* __restrict__ Amat,
                        const float* __restrict__ w1g, const float* __restrict__ b1g,
                        const float* __restrict__ w2g, const float* __restrict__ b2g,
                        const float* __restrict__ wpg, const float* __restrict__ bpg,
                        float* __restrict__ out)
{
    __shared__ float zsh[CCH * VP];
    __shared__ float k1s[CCH * VP];
    __shared__ float k2s[CCH * VP];
    __shared__ float k3s[CCH * VP];
    __shared__ float xin[CCH * VP];
    __shared__ float tmp[CCH * VP];
    __shared__ float pesh[CCH];
    __shared__ float bsh[3 * CCH];

    const int tid  = threadIdx.x;
    const int lane = tid & 31;
    const int wave = tid >> 5;
    const int mo   = wave >> 1;     // 0..3 : 16-row output tile
    const int no   = wave & 1;      // 0..1 : 16-col output tile
    const int koff = (lane >> 4) * 2;
    const int n = blockIdx.x;       // sample id = b*T + t
    const int b = n >> 6;
    const int t = n & 63;

    const size_t BCTV = (size_t)BBATCH * CCH * TT * VV;  // 6,553,600

    // ---- preload weight A-fragments into registers (amortized over 16 ode calls) ----
    const int Mrow = mo * 16 + (lane & 15);
    v2f w1f[16], w2f[16], wpf[16];
#pragma unroll
    for (int j = 0; j < 16; ++j) {
        const int k = 4 * j + koff;
        w1f[j].x = w1g[Mrow * 64 + k]; w1f[j].y = w1g[Mrow * 64 + k + 1];
        w2f[j].x = w2g[Mrow * 64 + k]; w2f[j].y = w2g[Mrow * 64 + k + 1];
        wpf[j].x = wpg[Mrow * 64 + k]; wpf[j].y = wpg[Mrow * 64 + k + 1];
    }

    // ---- preload adjacency B-fragments (B[k=u][n=v] = A[v][u]), zero-padded, into regs ----
    const int Ncol = no * 16 + (lane & 15);
    v2f aF[7];
#pragma unroll
    for (int k7 = 0; k7 < 7; ++k7) {
        const int kk = 4 * k7 + koff;
        aF[k7].x = (Ncol < VV && kk     < VV) ? Amat[Ncol * VV + kk]     : 0.0f;
        aF[k7].y = (Ncol < VV && kk + 1 < VV) ? Amat[Ncol * VV + kk + 1] : 0.0f;
    }

    // ---- stage biases and this sample's z into LDS ----
    if (tid < 64) {
        bsh[tid]       = b1g[tid];
        bsh[64 + tid]  = b2g[tid];
        bsh[128 + tid] = bpg[tid];
    }
    for (int idx = tid; idx < CCH * VP; idx += 256) {
        const int c = idx >> 5, v = idx & 31;
        zsh[idx] = (v < VV) ? z0[(((size_t)(b * CCH + c)) * TT + t) * VV + v] : 0.0f;
    }
    __syncthreads();

    // ---- slab 0: z_hat[0] = z0, z_cls[0] = z0 ----
    for (int idx = tid; idx < CCH * VP; idx += 256) {
        const int c = idx >> 5, v = idx & 31;
        if (v < VV) {
            const size_t e = (((size_t)(b * CCH + c)) * TT + t) * VV + v;
            const float val = zsh[idx];
            out[e] = val;               // z_hat slab 0
            out[5 * BCTV + e] = val;    // z_cls slab 0
        }
    }

    // ---- RK4 steps ----
    for (int step = 1; step <= NSTEP; ++step) {
#pragma unroll 1
        for (int sub = 0; sub < 4; ++sub) {   // t_off = {0,1,2,3}
            // positional encoding pe[(3t+sub), c]
            if (tid < 64) {
                const int i2 = tid >> 1;
                const float div = __expf(-(float)(2 * i2) * 0.14391156f); // ln(1e4)/64
                const float ang = (float)(3 * t + sub) * div;
                pesh[tid] = (tid & 1) ? __cosf(ang) : __sinf(ang);
            }
            __syncthreads();
            // build ode input: z + combo(k) + pe
            for (int idx = tid; idx < CCH * VP; idx += 256) {
                float kv = 0.0f;
                if (sub == 1)      kv = k1s[idx] * (1.0f / 3.0f);
                else if (sub == 2) kv = k2s[idx] - k1s[idx] * (1.0f / 3.0f);
                else if (sub == 3) kv = k1s[idx] - k2s[idx] + k3s[idx];
                xin[idx] = zsh[idx] + kv + pesh[idx >> 5];
            }
            __syncthreads();

            float* kdest = (sub == 0) ? k1s : (sub == 1) ? k2s : (sub == 2) ? k3s : zsh;
            // f_core: A, W1+relu, A, W2+relu, Wp
            amm(xin, aF, tmp, mo, no, lane, koff);                        __syncthreads();
            cmm(w1f, tmp, xin, bsh,        true,  mo, no, lane, koff);    __syncthreads();
            amm(xin, aF, tmp, mo, no, lane, koff);                        __syncthreads();
            cmm(w2f, tmp, xin, bsh + 64,   true,  mo, no, lane, koff);    __syncthreads();
            cmm(wpf, xin, kdest, bsh + 128, false, mo, no, lane, koff);   __syncthreads();
        }
        // z = (k1 + 3(k2+k3) + k4) / 8   (zsh currently holds k4)
        for (int idx = tid; idx < CCH * VP; idx += 256)
            zsh[idx] = (k1s[idx] + 3.0f * (k2s[idx] + k3s[idx]) + zsh[idx]) * 0.125f;
        __syncthreads();

        // ---- store slab `step`: z_cls direct, z_hat shifted by `step` ----
        const size_t hatbase = (size_t)step * BCTV;
        const size_t clsbase = 5 * BCTV + (size_t)step * BCTV;
        const int tp = t + step;  // DILATION = 1
        for (int idx = tid; idx < CCH * VP; idx += 256) {
            const int c = idx >> 5, v = idx & 31;
            if (v < VV) {
                const float val = zsh[idx];
                out[clsbase + (((size_t)(b * CCH + c)) * TT + t) * VV + v] = val;
                if (tp < TT)
                    out[hatbase + (((size_t)(b * CCH + c)) * TT + tp) * VV + v] = val;
                if (t < step)  // leading positions of shifted slab are zero
                    out[hatbase + (((size_t)(b * CCH + c)) * TT + t) * VV + v] = 0.0f;
            }
        }
        __syncthreads();
    }
}

extern "C" void kernel_launch(void* const* d_in, const int* in_sizes, int n_in,
                              void* d_out, int out_size, void* d_ws, size_t ws_size,
                              hipStream_t stream)
{
    (void)in_sizes; (void)n_in; (void)d_ws; (void)ws_size; (void)out_size;
    const float* z  = (const float*)d_in[0];
    const float* A  = (const float*)d_in[1];
    const float* w1 = (const float*)d_in[2];
    const float* b1 = (const float*)d_in[3];
    const float* w2 = (const float*)d_in[4];
    const float* b2 = (const float*)d_in[5];
    const float* wp = (const float*)d_in[6];
    const float* bp = (const float*)d_in[7];
    float* out = (float*)d_out;

    dim3 grid(BBATCH * TT);  // 4096 workgroups, one per (b,t) sample
    dim3 block(256);         // 8 wave32s -> 8 WMMA tiles per matmul
    hipLaunchKernelGGL(infogcn_rk4_kernel, grid, block, 0, stream,
                       z, A, w1, b1, w2, b2, wp, bp, out);
}